// UOTBadmmLoss_86311662780623
// MI455X (gfx1250) — compile-verified
//
#include <hip/hip_runtime.h>

// ===========================================================================
// UOT-BADMM contrastive loss, MI455X (gfx1250), collapsed-exact form.
//
// Derivation (bitwise-faithful to the reference, incl. torch broadcasting):
//   * z init = 0 and z <- z*(t - s) with finite (t - s)  =>  z == +/-0 exactly
//     for all 4 iterations (IEEE 0*finite = +/-0).
//   * Therefore y2 = z*log_t == +/-0, column max = +/-0, column sum = n exactly,
//     so log_s[i,j] = log_mu[j] - log(n): constant over rows (a vector).
//   * After iter 1 the reference's (1,N) axis-0 logsumexp is the identity,
//     so log_mu = log_eta = 0 exactly from iter 2 onward; z1/z2 are dead state.
//   * Unrolled: ls3 = -log n, log_eta3 = 0  =>
//       log_t_final = y4 - LSE_row(y4),  y4 = x - log n   (row log-softmax).
//     Iterations 1-3 contribute nothing to the output.
//   * loss = mean_i( LSE_j( exp(log_t_final)[i,:] ) - exp(log_t_final)[i,i] ).
//
// Pipeline: fp32 WMMA GEMM (V_WMMA_F32_16X16X4_F32, async double-buffered
// LDS tiles via GLOBAL_LOAD_ASYNC_TO_LDS_B128 / S_WAIT_ASYNCCNT) ->
// fused per-row softmax + cross-entropy (row resident in LDS).
// Traffic: ~134 MB total; X (67 MB) stays in the 192 MB L2 between kernels.
// ===========================================================================

typedef __attribute__((ext_vector_type(2))) float v2f;
typedef __attribute__((ext_vector_type(8))) float v8f;
typedef __attribute__((ext_vector_type(4))) int   v4i;

// Builtin param types (from hipcc diagnostic): global src is
// 'int __vector(4) __device__ *'  -> AS1 v4i*, LDS dst -> AS3 v4i*.
typedef __attribute__((address_space(1))) v4i* gbl_v4i_p;
typedef __attribute__((address_space(3))) v4i* lds_v4i_p;

#define TPB 256
#define NMAX 4096
#define FLT_NEG (-3.402823466e38f)

#if __has_builtin(__builtin_amdgcn_global_load_async_to_lds_b128) && \
    __has_builtin(__builtin_amdgcn_s_wait_asynccnt)
#define USE_ASYNC 1
#else
#define USE_ASYNC 0
#endif

// GEMM tiling: block tile 128x128, K tile 32; 256 threads = 8 waves (4M x 2N),
// each wave owns a 32x64 sub-tile = 2x4 grid of 16x16 fp32 WMMA accumulators.
#define BM 128
#define BN 128
#define BK 32
// LDS row stride 36 floats: 144 B rows -> every float4 chunk is 16 B aligned
// (async b128 dest) AND 36*m mod 64 is distinct for m=0..15 (conflict-free
// WMMA operand fetch).
#define LDSS 36

__device__ inline float blk_max(float v, float* red) {
  int tid = threadIdx.x;
  red[tid] = v; __syncthreads();
  for (int s = TPB >> 1; s > 0; s >>= 1) {
    if (tid < s) red[tid] = fmaxf(red[tid], red[tid + s]);
    __syncthreads();
  }
  float r = red[0]; __syncthreads();
  return r;
}

__device__ inline float blk_sum(float v, float* red) {
  int tid = threadIdx.x;
  red[tid] = v; __syncthreads();
  for (int s = TPB >> 1; s > 0; s >>= 1) {
    if (tid < s) red[tid] += red[tid + s];
    __syncthreads();
  }
  float r = red[0]; __syncthreads();
  return r;
}

// ---------------------------------------------------------------------------
// X = A @ B^T  (A,B: n x d fp32 row-major; X: n x n fp32)
// ---------------------------------------------------------------------------
__global__ __launch_bounds__(256) void gemm_xt_wmma(const float* __restrict__ A,
                                                    const float* __restrict__ B,
                                                    float* __restrict__ X,
                                                    int n, int d) {
  __shared__ __align__(16) float As[2][BM][LDSS];
  __shared__ __align__(16) float Bs[2][BN][LDSS];
  const int tid  = threadIdx.x;
  const int lane = tid & 31;
  const int wv   = tid >> 5;
  const int wm   = (wv & 3) << 5;    // 0,32,64,96
  const int wn   = (wv >> 2) << 6;   // 0,64
  const int m    = lane & 15;
  const int hi   = lane >> 4;
  const int row0 = blockIdx.y * BM;
  const int col0 = blockIdx.x * BN;
  // per-thread tile-load chunk coords: 4 float4 chunks each for A and B
  const int lr = tid >> 3;           // 0..31 (step 32 per chunk group)
  const int lc = (tid & 7) << 2;     // 0,4,...,28

  v8f acc[2][4];
#pragma unroll
  for (int mi = 0; mi < 2; ++mi)
#pragma unroll
    for (int ni = 0; ni < 4; ++ni) acc[mi][ni] = (v8f){};

#if USE_ASYNC
  // Issue one K-tile's loads: 8 async b128 per thread (=> 8 per wave, in-order
  // completion tracked by ASYNCcnt).
  auto issue = [&](int buf, int kb) {
#pragma unroll
    for (int i = 0; i < 4; ++i) {
      const float* ga = A + (size_t)(row0 + lr + 32 * i) * d + (kb + lc);
      __builtin_amdgcn_global_load_async_to_lds_b128(
          (gbl_v4i_p)ga, (lds_v4i_p)&As[buf][lr + 32 * i][lc], 0, 0);
    }
#pragma unroll
    for (int i = 0; i < 4; ++i) {
      const float* gb = B + (size_t)(col0 + lr + 32 * i) * d + (kb + lc);
      __builtin_amdgcn_global_load_async_to_lds_b128(
          (gbl_v4i_p)gb, (lds_v4i_p)&Bs[buf][lr + 32 * i][lc], 0, 0);
    }
  };
#else
  auto issue = [&](int buf, int kb) {
#pragma unroll
    for (int i = 0; i < 4; ++i) {
      float4 va = *(const float4*)(A + (size_t)(row0 + lr + 32 * i) * d + (kb + lc));
      *(float4*)&As[buf][lr + 32 * i][lc] = va;
      float4 vb = *(const float4*)(B + (size_t)(col0 + lr + 32 * i) * d + (kb + lc));
      *(float4*)&Bs[buf][lr + 32 * i][lc] = vb;
    }
  };
#endif

  auto compute = [&](int buf) {
#pragma unroll
    for (int kk = 0; kk < BK; kk += 4) {
      v2f a[2], b[4];
#pragma unroll
      for (int mi = 0; mi < 2; ++mi) {
        // A operand: lane l holds A[m = l&15][kk + 2*(l>>4) + r], r = 0,1
        a[mi][0] = As[buf][wm + 16 * mi + m][kk + (hi << 1)];
        a[mi][1] = As[buf][wm + 16 * mi + m][kk + (hi << 1) + 1];
      }
#pragma unroll
      for (int ni = 0; ni < 4; ++ni) {
        // B operand: lane l holds Bmat[k = 2r + (l>>4)][n = l&15]
        b[ni][0] = Bs[buf][wn + 16 * ni + m][kk + hi];
        b[ni][1] = Bs[buf][wn + 16 * ni + m][kk + 2 + hi];
      }
#pragma unroll
      for (int mi = 0; mi < 2; ++mi)
#pragma unroll
        for (int ni = 0; ni < 4; ++ni)
          acc[mi][ni] = __builtin_amdgcn_wmma_f32_16x16x4_f32(
              false, a[mi], false, b[ni], (short)0, acc[mi][ni], false, false);
    }
  };

  const int nk = d / BK;
  issue(0, 0);
  for (int kbi = 0; kbi < nk; ++kbi) {
    const int buf = kbi & 1;
    if (kbi + 1 < nk) {
      issue(buf ^ 1, (kbi + 1) * BK);          // prefetch next tile
#if USE_ASYNC
      __builtin_amdgcn_s_wait_asynccnt(8);     // current tile's 8 ops done
#endif
    } else {
#if USE_ASYNC
      __builtin_amdgcn_s_wait_asynccnt(0);
#endif
    }
    __syncthreads();   // all waves' slices of tile `buf` landed in LDS
    compute(buf);
    __syncthreads();   // done reading `buf` before it is overwritten
  }

  // C/D layout: VGPR r, lanes 0-15: (M=r, N=lane); lanes 16-31: (M=8+r, N=lane-16)
#pragma unroll
  for (int mi = 0; mi < 2; ++mi)
#pragma unroll
    for (int ni = 0; ni < 4; ++ni) {
      float* out = X + (size_t)(row0 + wm + 16 * mi + (hi << 3)) * n
                     + col0 + wn + 16 * ni + m;
#pragma unroll
      for (int r = 0; r < 8; ++r) out[(size_t)r * n] = acc[mi][ni][r];
    }
}

// ---------------------------------------------------------------------------
__global__ void zero_out(float* __restrict__ out) { out[0] = 0.f; }

// ---------------------------------------------------------------------------
// Per-row: y = x - log n; lse = LSE(y); p = exp(y - lse)   (== t matrix)
//          loss_i = (max(p) + log sum exp(p - max(p))) - p[i]
// One workgroup per row; row lives in LDS (16 KB of 320 KB/WGP).
// ---------------------------------------------------------------------------
__global__ __launch_bounds__(256) void softmax_ce_rows(const float* __restrict__ X,
                                                       float* __restrict__ out,
                                                       float logn, int n) {
  __shared__ __align__(16) float ys[NMAX];
  __shared__ float red[TPB];
  const int i = blockIdx.x;
  const int tid = threadIdx.x;
  const float4* X4 = (const float4*)(X + (size_t)i * n);
  float4* ys4 = (float4*)ys;

  float mx = FLT_NEG;
  for (int j4 = tid; j4 < (n >> 2); j4 += TPB) {
    float4 x = X4[j4];
    float4 y = make_float4(x.x - logn, x.y - logn, x.z - logn, x.w - logn);
    ys4[j4] = y;
    mx = fmaxf(mx, fmaxf(fmaxf(y.x, y.y), fmaxf(y.z, y.w)));
  }
  __syncthreads();
  mx = blk_max(mx, red);
  float sm = 0.f;
  for (int j = tid; j < n; j += TPB) sm += expf(ys[j] - mx);
  sm = blk_sum(sm, red);
  const float lse = mx + logf(sm);

  // p = exp(y - lse): these are the "logits" fed to cross-entropy
  float mx2 = FLT_NEG;
  for (int j = tid; j < n; j += TPB) {
    float p = expf(ys[j] - lse);
    ys[j] = p;
    mx2 = fmaxf(mx2, p);
  }
  __syncthreads();
  mx2 = blk_max(mx2, red);
  float sm2 = 0.f;
  for (int j = tid; j < n; j += TPB) sm2 += expf(ys[j] - mx2);
  sm2 = blk_sum(sm2, red);

  if (tid == 0) {
    float lse2 = mx2 + logf(sm2);
    atomicAdd(out, (lse2 - ys[i]) / (float)n);   // labels = arange -> diagonal
  }
}

// ---------------------------------------------------------------------------
extern "C" void kernel_launch(void* const* d_in, const int* in_sizes, int n_in,
                              void* d_out, int out_size, void* d_ws, size_t ws_size,
                              hipStream_t stream) {
  const float* img = (const float*)d_in[0];   // (n, d) fp32
  const float* txt = (const float*)d_in[1];   // (n, d) fp32
  // d_in[2] logit_scale unused by the forward; d_in[3] labels = arange
  const int n = in_sizes[3];
  const int d = in_sizes[0] / n;

  float* X = (float*)d_ws;                    // (n, n) logits, 67 MB
  float* out = (float*)d_out;

  dim3 gg(n / BN, n / BM);
  gemm_xt_wmma<<<gg, 256, 0, stream>>>(img, txt, X, n, d);
  zero_out<<<1, 1, 0, stream>>>(out);
  softmax_ce_rows<<<n, TPB, 0, stream>>>(X, out, logf((float)n), n);
}